// SimpleAttention_58531814310125
// MI455X (gfx1250) — compile-verified
//
#include <hip/hip_runtime.h>
#include <hip/hip_bf16.h>
#include <math.h>

#define NUM_HEADS 32
#define HEAD_DIM  128
#define QLORA     1536
#define BSZ       8
#define SEQ_LEN   4096
#define QOUT      (NUM_HEADS * HEAD_DIM)      // 4096
#define SPLITS    4
#define SEQ_PER_SPLIT (SEQ_LEN / SPLITS)      // 1024
#define T2        256                         // threads per attention block
#define S_PER_T   (SEQ_PER_SPLIT / T2)        // 4 seq positions per thread

typedef __attribute__((ext_vector_type(2))) float v2f;
typedef __attribute__((ext_vector_type(8))) float v8f;

// ---------------------------------------------------------------------------
// Kernel 1: q = hidden_states_q @ W_q + b_q   (M=8 (pad 16), K=1536, N=4096)
// V_WMMA_F32_16X16X4_F32. A/B lane layout (ISA 7.12.2): element
// [m_or_n = lane%16, k = vgpr + (lane/16)*2]. D: row = vgpr + (lane/16)*8.
// Batch rows 8..15 alias row 0 (valid memory); their D rows are never stored,
// and D[m,*] depends only on A row m, so no masking is needed -> no extra
// VALU op feeding each WMMA.
// One wave per 16-wide N tile; 8 waves/block, 32 blocks.
// ---------------------------------------------------------------------------
__global__ __launch_bounds__(256) void qproj_wmma(const float* __restrict__ hq,
                                                  const float* __restrict__ Wq,
                                                  const float* __restrict__ bq,
                                                  float* __restrict__ qout) {
  const int lane = threadIdx.x & 31;
  const int wave = threadIdx.x >> 5;
  const int n0   = (blockIdx.x * 8 + wave) * 16;
  const int half = lane >> 4;     // which K-half of the 4-wide chunk
  const int mn   = lane & 15;     // A row (m) / B column (n)

  const float* arow = hq + (size_t)(mn < BSZ ? mn : 0) * QLORA;
  const float* bcol = Wq + n0 + mn;

  v8f c = {0.f, 0.f, 0.f, 0.f, 0.f, 0.f, 0.f, 0.f};
  for (int k0 = 0; k0 < QLORA; k0 += 4) {
    const int k = k0 + half * 2;
    v2f a, b;
    a.x = arow[k];                         // contiguous pair -> global_load_b64
    a.y = arow[k + 1];
    b.x = bcol[(size_t)k       * QOUT];
    b.y = bcol[(size_t)(k + 1) * QOUT];
    // D = A(16x4) * B(4x16) + C ; emits v_wmma_f32_16x16x4_f32
    c = __builtin_amdgcn_wmma_f32_16x16x4_f32(false, a, false, b, (short)0, c,
                                              false, false);
  }

  if (half == 0) {                         // lanes 0..15 hold D rows 0..7 (vgpr r)
    const float bias = bq[n0 + mn];
#pragma unroll
    for (int r = 0; r < BSZ; ++r)
      qout[(size_t)r * QOUT + n0 + mn] = c[r] + bias;
  }
}

// ---------------------------------------------------------------------------
// Kernel 2: flash-decode partial attention. One (b, h, split) per block.
// Each thread owns one seq position per iteration: streams its 1 KB K|V row
// once with float4 loads, online softmax, 128-float register accumulator.
// Block combine via LDS tree reductions + ds_add_f32 atomics.
// ---------------------------------------------------------------------------
__global__ __launch_bounds__(256) void attn_partial(const float* __restrict__ kv,
                                                    const float* __restrict__ qws,
                                                    float* __restrict__ pml,
                                                    float* __restrict__ pacc) {
  const int split = blockIdx.x & (SPLITS - 1);
  const int h     = (blockIdx.x >> 2) & (NUM_HEADS - 1);
  const int b     = blockIdx.x >> 7;          // / (SPLITS*NUM_HEADS) = /128
  const int t     = threadIdx.x;

  __shared__ float sq[HEAD_DIM];
  __shared__ float sred[T2];
  __shared__ float sacc[HEAD_DIM];

  if (t < HEAD_DIM) {
    sq[t]   = qws[(size_t)b * QOUT + h * HEAD_DIM + t];
    sacc[t] = 0.f;
  }
  __syncthreads();

  float  m = -INFINITY, l = 0.f;
  float4 acc[HEAD_DIM / 4];
#pragma unroll
  for (int d = 0; d < HEAD_DIM / 4; ++d) acc[d] = make_float4(0.f, 0.f, 0.f, 0.f);

  for (int it = 0; it < S_PER_T; ++it) {
    const int    s  = split * SEQ_PER_SPLIT + it * T2 + t;
    const float* kp = kv + (((size_t)b * SEQ_LEN + s) * NUM_HEADS + h) * (2 * HEAD_DIM);
    const float* vp = kp + HEAD_DIM;
    __builtin_prefetch(vp, 0, 1);            // gfx1250 global_prefetch_b8

    float sc = 0.f;
#pragma unroll
    for (int d = 0; d < HEAD_DIM / 4; ++d) {
      const float4 k4 = ((const float4*)kp)[d];
      sc = fmaf(sq[4 * d + 0], k4.x, sc);
      sc = fmaf(sq[4 * d + 1], k4.y, sc);
      sc = fmaf(sq[4 * d + 2], k4.z, sc);
      sc = fmaf(sq[4 * d + 3], k4.w, sc);
    }

    const float mnew  = fmaxf(m, sc);
    const float scale = __expf(m - mnew);    // first iter: exp(-inf)=0
    const float p     = __expf(sc - mnew);
    l = fmaf(l, scale, p);
#pragma unroll
    for (int d = 0; d < HEAD_DIM / 4; ++d) {
      const float4 v4 = ((const float4*)vp)[d];
      acc[d].x = fmaf(acc[d].x, scale, p * v4.x);
      acc[d].y = fmaf(acc[d].y, scale, p * v4.y);
      acc[d].z = fmaf(acc[d].z, scale, p * v4.z);
      acc[d].w = fmaf(acc[d].w, scale, p * v4.w);
    }
    m = mnew;
  }

  // block max of m
  sred[t] = m;
  __syncthreads();
  for (int o = T2 / 2; o > 0; o >>= 1) {
    if (t < o) sred[t] = fmaxf(sred[t], sred[t + o]);
    __syncthreads();
  }
  const float M = sred[0];
  __syncthreads();

  // block sum of rescaled l
  const float f = __expf(m - M);
  sred[t] = l * f;
  __syncthreads();
  for (int o = T2 / 2; o > 0; o >>= 1) {
    if (t < o) sred[t] += sred[t + o];
    __syncthreads();
  }
  const float L = sred[0];

  // combine per-thread accumulators (ds_add_f32 atomics)
#pragma unroll
  for (int d = 0; d < HEAD_DIM / 4; ++d) {
    atomicAdd(&sacc[4 * d + 0], f * acc[d].x);
    atomicAdd(&sacc[4 * d + 1], f * acc[d].y);
    atomicAdd(&sacc[4 * d + 2], f * acc[d].z);
    atomicAdd(&sacc[4 * d + 3], f * acc[d].w);
  }
  __syncthreads();

  const int bhs = (b * NUM_HEADS + h) * SPLITS + split;
  if (t == 0) {
    pml[bhs * 2 + 0] = M;
    pml[bhs * 2 + 1] = L;
  }
  if (t < HEAD_DIM) pacc[(size_t)bhs * HEAD_DIM + t] = sacc[t];
}

// ---------------------------------------------------------------------------
// Kernel 3: merge the SPLITS partials per (b,h) and normalize.
// ---------------------------------------------------------------------------
__global__ __launch_bounds__(128) void attn_merge(const float* __restrict__ pml,
                                                  const float* __restrict__ pacc,
                                                  float* __restrict__ out) {
  const int bh = blockIdx.x;
  const int d  = threadIdx.x;

  float M = -INFINITY;
#pragma unroll
  for (int i = 0; i < SPLITS; ++i) M = fmaxf(M, pml[(bh * SPLITS + i) * 2]);

  float L = 0.f, a = 0.f;
#pragma unroll
  for (int i = 0; i < SPLITS; ++i) {
    const float fi = __expf(pml[(bh * SPLITS + i) * 2] - M);
    L += pml[(bh * SPLITS + i) * 2 + 1] * fi;
    a += fi * pacc[(size_t)(bh * SPLITS + i) * HEAD_DIM + d];
  }
  out[(size_t)bh * HEAD_DIM + d] = a / L;
}

// ---------------------------------------------------------------------------
extern "C" void kernel_launch(void* const* d_in, const int* in_sizes, int n_in,
                              void* d_out, int out_size, void* d_ws, size_t ws_size,
                              hipStream_t stream) {
  const float* hq = (const float*)d_in[0];   // (8, 1536)
  const float* kv = (const float*)d_in[1];   // (8, 4096, 32, 256)
  const float* Wq = (const float*)d_in[2];   // (1536, 4096)
  const float* bq = (const float*)d_in[3];   // (4096,)
  float*       out = (float*)d_out;          // (8, 4096)

  float* qws  = (float*)d_ws;                               // 8*4096 floats
  float* pml  = qws + (size_t)BSZ * QOUT;                   // 256*SPLITS*2 floats
  float* pacc = pml + (size_t)BSZ * NUM_HEADS * SPLITS * 2; // 256*SPLITS*128 floats

  hipLaunchKernelGGL(qproj_wmma, dim3(QOUT / (16 * 8)), dim3(256), 0, stream,
                     hq, Wq, bq, qws);
  hipLaunchKernelGGL(attn_partial, dim3(BSZ * NUM_HEADS * SPLITS), dim3(T2), 0,
                     stream, kv, qws, pml, pacc);
  hipLaunchKernelGGL(attn_merge, dim3(BSZ * NUM_HEADS), dim3(HEAD_DIM), 0,
                     stream, pml, pacc, out);
}